// HD_loss_2336462209299
// MI455X (gfx1250) — compile-verified
//
#include <hip/hip_runtime.h>

// Problem constants (pred: (2,2,96,96,96) f32, target: (2,96,96,96) int)
#define NX      96
#define NYZ     9216        // 96*96
#define VOL     884736      // 96^3
#define NVOL    8           // {pred_fg,pred_bg,targ_fg,targ_bg} x {b=0,1}
#define NTOT    1769472     // 2 * VOL  (loss elements / mask elements)
#define BIGF    1e10f
#define NBLK    1024

typedef float v2f __attribute__((ext_vector_type(2)));
typedef float v8f __attribute__((ext_vector_type(8)));

#if __has_builtin(__builtin_amdgcn_global_load_async_to_lds_b32) && \
    __has_builtin(__builtin_amdgcn_s_wait_asynccnt)
#define HAVE_ASYNC_LDS 1
typedef __attribute__((address_space(1))) int as1_int;   // builtin wants non-const
typedef __attribute__((address_space(3))) int as3_int;
#endif

// ---------------------------------------------------------------- flags zero
__global__ void k_zero_flags(int* flags) {
    if (threadIdx.x < 8) flags[threadIdx.x] = 0;
}

// ------------------------------------------------- masks -> EDT init fields
// f[(field*2+b)*VOL + idx]; field 0=pred_fg 1=pred_bg 2=targ_fg 3=targ_bg
__global__ void k_mask_init(const float* __restrict__ pred,
                            const int*   __restrict__ target,
                            float* __restrict__ f,
                            int*   __restrict__ flags) {
    int stride = gridDim.x * blockDim.x;
    for (int g = blockIdx.x * blockDim.x + threadIdx.x; g < NTOT; g += stride) {
        int b   = g / VOL;
        int idx = g - b * VOL;
        float p0 = pred[(b * 2 + 0) * VOL + idx];
        float p1 = pred[(b * 2 + 1) * VOL + idx];
        bool pm = p1 > p0;              // argmax==1 (ties -> channel 0)
        bool tm = (target[g] == 1);
        f[(0 * 2 + b) * VOL + idx] = pm ? BIGF : 0.0f;
        f[(1 * 2 + b) * VOL + idx] = pm ? 0.0f : BIGF;
        f[(2 * 2 + b) * VOL + idx] = tm ? BIGF : 0.0f;
        f[(3 * 2 + b) * VOL + idx] = tm ? 0.0f : BIGF;
        // any_fg flags (VOL divisible by 32 -> warp never straddles b)
        unsigned long long bp = __ballot(pm);
        unsigned long long bt = __ballot(tm);
        if ((threadIdx.x & 31) == 0) {
            if (bp) atomicOr(&flags[b],     1);
            if (bt) atomicOr(&flags[2 + b], 1);
        }
    }
}

// ------------------------------------- exact 1D min-plus EDT pass, in place
// one 96-thread block per line; D[i] = min_j line[j] + (i-j)^2  (fp32, exact
// match to the reference brute-force jnp.min(f + cost)).  Line staging uses
// GLOBAL_LOAD_ASYNC_TO_LDS (ASYNCcnt) when the toolchain exposes it.
__global__ void k_dt1d(float* __restrict__ f, int strideA, int strideC, int strideI) {
    __shared__ float line[NX];
    int L   = blockIdx.x;               // 0 .. NVOL*9216-1
    int vol = L / NYZ;
    int r   = L - vol * NYZ;
    int a   = r / NX;
    int c   = r - a * NX;
    long base = (long)vol * VOL + (long)a * strideA + (long)c * strideC;
    int i = threadIdx.x;

#ifdef HAVE_ASYNC_LDS
    {
        // generic->AS1 is value-identical; generic->AS3 truncates to the
        // 32-bit LDS offset (AMDGPU aperture layout).
        const float* gp = f + base + (long)i * strideI;
        as1_int* gsrc = (as1_int*)(unsigned long long)(size_t)gp;
        as3_int* ldst = (as3_int*)(unsigned int)(unsigned long long)(size_t)&line[i];
        __builtin_amdgcn_global_load_async_to_lds_b32(gsrc, ldst, 0, 0);
        __builtin_amdgcn_s_wait_asynccnt(0);   // own async load complete
    }
#else
    line[i] = f[base + (long)i * strideI];
#endif
    __syncthreads();                           // all waves' loads complete

    float fi = (float)i;
    float d  = line[0] + fi * fi;
#pragma unroll 8
    for (int j = 1; j < NX; ++j) {
        float dj = fi - (float)j;
        d = fminf(d, line[j] + dj * dj);
    }
    f[base + (long)i * strideI] = d;
}

// ------------------------------------------------------------ loss element
__device__ __forceinline__ float loss_elem(long g,
                                           const float* __restrict__ pred,
                                           const int*   __restrict__ target,
                                           const float* __restrict__ f,
                                           const int*   __restrict__ flags) {
    long gg = (g < NTOT) ? g : 0;       // clamp: keep control flow uniform
    int  b   = (int)(gg / VOL);
    int  idx = (int)(gg - (long)b * VOL);
    float p1  = pred[(b * 2 + 1) * VOL + idx];
    float toh = (target[gg] == 1) ? 1.0f : 0.0f;
    float e   = p1 - toh; e = e * e;
    float pfg = f[(0 * 2 + b) * VOL + idx];
    float pbg = f[(1 * 2 + b) * VOL + idx];
    float tfg = f[(2 * 2 + b) * VOL + idx];
    float tbg = f[(3 * 2 + b) * VOL + idx];
    float pdt = sqrtf(pfg) + sqrtf(pbg);
    float tdt = sqrtf(tfg) + sqrtf(tbg);
    float pd  = flags[b]     ? pdt * pdt : 0.0f;   // ALPHA == 2
    float td  = flags[2 + b] ? tdt * tdt : 0.0f;
    float res = e * (pd + td);
    return (g < NTOT) ? res : 0.0f;
}

// --------------------------- WMMA-accumulated weighted-sum block reduction
// B = ones  =>  D[m,n] = sum_k A[m,k]; feeding 64 products per chain-iter
// into A and chaining C makes v_wmma_f32_16x16x4_f32 a 64-wide f32 adder.
// Two independent accumulator chains overlap WMMA latency.  Every product is
// replicated across the 16 columns -> divide block sum by 16.
__global__ void k_reduce_wmma(const float* __restrict__ pred,
                              const int*   __restrict__ target,
                              const float* __restrict__ f,
                              const int*   __restrict__ flags,
                              float* __restrict__ partials) {
    __shared__ float red[256];
    int lane   = threadIdx.x & 31;
    int nwaves = (gridDim.x * blockDim.x) >> 5;
    int wave   = (blockIdx.x * (blockDim.x >> 5)) + (threadIdx.x >> 5);

    v8f acc0 = {};
    v8f acc1 = {};
    v2f ones; ones.x = 1.0f; ones.y = 1.0f;

    for (long base = (long)wave * 128; base < NTOT; base += (long)nwaves * 128) {
        float a0 = loss_elem(base + lane,       pred, target, f, flags);
        float a1 = loss_elem(base + 32 + lane,  pred, target, f, flags);
        float b0 = loss_elem(base + 64 + lane,  pred, target, f, flags);
        float b1 = loss_elem(base + 96 + lane,  pred, target, f, flags);
        v2f A0; A0.x = a0; A0.y = a1;
        v2f A1; A1.x = b0; A1.y = b1;
        // 8 args: (neg_a, A, neg_b, B, c_mod, C, reuse_a, reuse_b)
        acc0 = __builtin_amdgcn_wmma_f32_16x16x4_f32(
            false, A0, false, ones, (short)0, acc0, false, false);
        acc1 = __builtin_amdgcn_wmma_f32_16x16x4_f32(
            false, A1, false, ones, (short)0, acc1, false, false);
    }
    float s = (acc0[0] + acc1[0]) + (acc0[1] + acc1[1]) +
              (acc0[2] + acc1[2]) + (acc0[3] + acc1[3]) +
              (acc0[4] + acc1[4]) + (acc0[5] + acc1[5]) +
              (acc0[6] + acc1[6]) + (acc0[7] + acc1[7]);
    red[threadIdx.x] = s;
    __syncthreads();
    for (int off = 128; off > 0; off >>= 1) {
        if (threadIdx.x < off) red[threadIdx.x] += red[threadIdx.x + off];
        __syncthreads();
    }
    if (threadIdx.x == 0) partials[blockIdx.x] = red[0] * (1.0f / 16.0f);
}

// ------------------------------------------------------------------ finalize
__global__ void k_finalize(const float* __restrict__ partials, float* __restrict__ out) {
    __shared__ float red[256];
    float s = 0.0f;
    for (int i = threadIdx.x; i < NBLK; i += 256) s += partials[i];
    red[threadIdx.x] = s;
    __syncthreads();
    for (int off = 128; off > 0; off >>= 1) {
        if (threadIdx.x < off) red[threadIdx.x] += red[threadIdx.x + off];
        __syncthreads();
    }
    if (threadIdx.x == 0) out[0] = red[0] / (float)NTOT;
}

// -------------------------------------------------------------------- launch
extern "C" void kernel_launch(void* const* d_in, const int* in_sizes, int n_in,
                              void* d_out, int out_size, void* d_ws, size_t ws_size,
                              hipStream_t stream) {
    const float* pred   = (const float*)d_in[0];   // (2,2,96,96,96) f32
    const int*   target = (const int*)d_in[1];     // (2,96,96,96)
    float* fvol    = (float*)d_ws;                                   // 8*VOL f32
    int*   flags   = (int*)((char*)d_ws + (size_t)NVOL * VOL * 4);   // 8 ints
    float* partial = (float*)(flags + 8);                            // NBLK f32
    float* out     = (float*)d_out;

    k_zero_flags<<<1, 32, 0, stream>>>(flags);
    k_mask_init <<<1024, 256, 0, stream>>>(pred, target, fvol, flags);
    // separable EDT, axis order X,Y,Z as in the reference
    k_dt1d<<<NVOL * NYZ, NX, 0, stream>>>(fvol,   96,  1, NYZ); // X lines (a=y,c=z)
    k_dt1d<<<NVOL * NYZ, NX, 0, stream>>>(fvol,  NYZ,  1,  96); // Y lines (a=x,c=z)
    k_dt1d<<<NVOL * NYZ, NX, 0, stream>>>(fvol,  NYZ, 96,   1); // Z lines (a=x,c=y)
    k_reduce_wmma<<<NBLK, 256, 0, stream>>>(pred, target, fvol, flags, partial);
    k_finalize<<<1, 256, 0, stream>>>(partial, out);
}